// Model_38603166056697
// MI455X (gfx1250) — compile-verified
//
#include <hip/hip_runtime.h>
#include <hip/hip_bf16.h>

typedef __attribute__((ext_vector_type(2))) float v2f;
typedef __attribute__((ext_vector_type(8))) float v8f;

static inline int cdiv_host(long a, long b) { return (int)((a + b - 1) / b); }

// ---------------------------------------------------------------------------
// Degree / normalization
// ---------------------------------------------------------------------------
__global__ void deg_init_kernel(float* __restrict__ deg, int n) {
    int i = blockIdx.x * blockDim.x + threadIdx.x;
    if (i < n) deg[i] = 1.0f;   // self loop
}

__global__ void deg_accum_kernel(const int* __restrict__ dst, float* __restrict__ deg, int e) {
    int i = blockIdx.x * blockDim.x + threadIdx.x;
    if (i < e) atomicAdd(&deg[dst[i]], 1.0f);
}

__global__ void deg_rsqrt_kernel(float* __restrict__ deg, int n) {
    int i = blockIdx.x * blockDim.x + threadIdx.x;
    if (i < n) deg[i] = rsqrtf(deg[i]);   // deg >= 1 always
}

// ---------------------------------------------------------------------------
// Dense transform: Y[n, DOUT] = X[n, K] @ W[K, DOUT] using fp32 WMMA 16x16x4.
// 256 threads = 8 waves; each wave computes one 16x16 output tile.
// W (K*DOUT <= 8192 floats) and the X row-tile are staged in LDS.
// ---------------------------------------------------------------------------
template<int K, int DOUT>
__global__ __launch_bounds__(256) void gemm_wmma_kernel(const float* __restrict__ X,
                                                        const float* __restrict__ W,
                                                        float* __restrict__ Y, int n) {
    constexpr int ROWS = 2048 / DOUT;     // 32 (DOUT=64) or 64 (DOUT=32)
    constexpr int COLT = DOUT / 16;       // column tiles per block
    __shared__ float Wlds[K * DOUT];
    __shared__ float Xlds[ROWS * K];

    const int t = threadIdx.x;

    // cooperative load of W into LDS (float4, fully coalesced)
    for (int i = t; i < (K * DOUT) / 4; i += 256) {
        ((float4*)Wlds)[i] = ((const float4*)W)[i];
    }
    // cooperative load of the contiguous X row-tile into LDS
    const long rowBase = (long)blockIdx.x * ROWS;
    const long totalElems = (long)n * K;
    for (int i = t; i < (ROWS * K) / 4; i += 256) {
        long gidx = rowBase * K + (long)i * 4;
        float4 v = make_float4(0.f, 0.f, 0.f, 0.f);
        if (gidx + 3 < totalElems) v = ((const float4*)X)[rowBase * K / 4 + i];
        ((float4*)Xlds)[i] = v;
    }
    __syncthreads();

    const int wave = t >> 5;
    const int lane = t & 31;
    const int half = lane >> 4;       // 0: K pair {k,k+1}, 1: {k+2,k+3}
    const int l16  = lane & 15;
    const int rowTile = wave / COLT;
    const int colTile = wave % COLT;
    const int r0 = rowTile * 16;      // row offset inside block tile
    const int c0 = colTile * 16;      // output column offset

    v8f acc = {};
    #pragma unroll 4
    for (int k = 0; k < K; k += 4) {
        const int kk = k + 2 * half;
        v2f a, b;
        // A 16x4 fp32: lanes 0-15 hold M=0..15 K={k,k+1}; lanes 16-31 K={k+2,k+3}
        a.x = Xlds[(r0 + l16) * K + kk];
        a.y = Xlds[(r0 + l16) * K + kk + 1];
        // B 4x16 fp32: lanes 0-15 hold N=0..15 K={k,k+1}; lanes 16-31 K={k+2,k+3}
        b.x = Wlds[kk * DOUT + c0 + l16];
        b.y = Wlds[(kk + 1) * DOUT + c0 + l16];
        acc = __builtin_amdgcn_wmma_f32_16x16x4_f32(
            /*neg_a=*/false, a, /*neg_b=*/false, b,
            /*c_mod=*/(short)0, acc, /*reuse_a=*/false, /*reuse_b=*/false);
    }

    // C/D layout: VGPR r, lanes 0-15 -> M=r, N=lane; lanes 16-31 -> M=8+r, N=lane-16
    const long gRow0 = rowBase + r0 + 8 * half;
    float* yBase = Y + gRow0 * DOUT + c0 + l16;
    if (rowBase + ROWS <= (long)n) {
        // full tile: block-uniform branch, no per-row exec masking
        #pragma unroll
        for (int r = 0; r < 8; ++r) yBase[(long)r * DOUT] = acc[r];
    } else {
        #pragma unroll
        for (int r = 0; r < 8; ++r) {
            if (gRow0 + r < n) yBase[(long)r * DOUT] = acc[r];
        }
    }
}

// ---------------------------------------------------------------------------
// agg[n,c] = bias[c] + H[n,c] * dinv[n]^2   (self-loop term folded in)
// node index is wave-uniform -> scalar dinv load
// ---------------------------------------------------------------------------
template<int D>
__global__ void agg_init_kernel(const float* __restrict__ H, const float* __restrict__ dinv,
                                const float* __restrict__ bias, float* __restrict__ agg, int n) {
    long i = (long)blockIdx.x * blockDim.x + threadIdx.x;
    if (i >= (long)n * D) return;
    int node = __builtin_amdgcn_readfirstlane((int)(i / D));  // uniform within wave
    int c = (int)(i % D);
    float di = dinv[node];
    agg[i] = bias[c] + H[i] * di * di;
}

// ---------------------------------------------------------------------------
// agg[dst, c] += H[src, c] * dinv[src]*dinv[dst] for every edge.
// Lane = channel; edge id is wave-uniform -> src/dst/dinv become scalar loads,
// so VMEM is spent only on the coalesced H gather + float atomics.
// ---------------------------------------------------------------------------
template<int D>
__global__ void edge_scatter_kernel(const float* __restrict__ H, const int* __restrict__ src,
                                    const int* __restrict__ dst, const float* __restrict__ dinv,
                                    float* __restrict__ agg, int e) {
    long i = (long)blockIdx.x * blockDim.x + threadIdx.x;
    if (i >= (long)e * D) return;
    int ed = __builtin_amdgcn_readfirstlane((int)(i / D));    // uniform within wave
    int c  = (int)(i % D);
    int s = src[ed];
    int d = dst[ed];
    float nrm = dinv[s] * dinv[d];
    atomicAdd(&agg[(long)d * D + c], H[(long)s * D + c] * nrm);
}

// ---------------------------------------------------------------------------
// BatchNorm: per-channel sum / sumsq with block-local LDS reduction
// ---------------------------------------------------------------------------
__global__ void bn_zero_kernel(float* __restrict__ p, int k) {
    int i = blockIdx.x * blockDim.x + threadIdx.x;
    if (i < k) p[i] = 0.0f;
}

template<int D>
__global__ __launch_bounds__(256) void bn_stats_kernel(const float* __restrict__ X,
                                                       float* __restrict__ sums, int n) {
    constexpr int RL = 256 / D;           // row-lanes per block
    __shared__ float sh[512];
    const int t = threadIdx.x;
    const int c = t % D;
    const int rl = t / D;
    float s = 0.f, q = 0.f;
    for (long row = (long)blockIdx.x * RL + rl; row < n; row += (long)gridDim.x * RL) {
        float v = X[row * D + c];
        s += v;
        q += v * v;
    }
    sh[t] = s;
    sh[256 + t] = q;
    __syncthreads();
    if (rl == 0) {
        #pragma unroll
        for (int j = 1; j < RL; ++j) {
            s += sh[j * D + c];
            q += sh[256 + j * D + c];
        }
        atomicAdd(&sums[c], s);
        atomicAdd(&sums[D + c], q);
    }
}

template<int D>
__global__ void bn_final_kernel(const float* __restrict__ sums, const float* __restrict__ gamma,
                                const float* __restrict__ beta, float* __restrict__ ab, float inv_n) {
    int c = threadIdx.x;
    if (c < D) {
        float mu  = sums[c] * inv_n;
        float var = sums[D + c] * inv_n - mu * mu;
        float a   = gamma[c] * rsqrtf(var + 1e-5f);
        ab[c]     = a;
        ab[D + c] = beta[c] - mu * a;
    }
}

template<int D, bool RELU, bool NT>
__global__ void bn_apply_kernel(const float* __restrict__ X, const float* __restrict__ ab,
                                float* __restrict__ Y, int n) {
    long i = (long)blockIdx.x * blockDim.x + threadIdx.x;
    if (i >= (long)n * D) return;
    int c = (int)(i % D);
    float v = ab[c] * X[i] + ab[D + c];
    if (RELU) v = fmaxf(v, 0.f);
    if (NT) __builtin_nontemporal_store(v, &Y[i]);   // final output: never re-read
    else    Y[i] = v;
}

// ---------------------------------------------------------------------------
// Host orchestration
// ---------------------------------------------------------------------------
extern "C" void kernel_launch(void* const* d_in, const int* in_sizes, int n_in,
                              void* d_out, int out_size, void* d_ws, size_t ws_size,
                              hipStream_t stream) {
    const float* x   = (const float*)d_in[0];
    const int*   ei  = (const int*)d_in[1];
    const float* W1  = (const float*)d_in[2];
    const float* b1  = (const float*)d_in[3];
    const float* g1  = (const float*)d_in[4];
    const float* be1 = (const float*)d_in[5];
    const float* W2  = (const float*)d_in[6];
    const float* b2  = (const float*)d_in[7];
    const float* g2  = (const float*)d_in[8];
    const float* be2 = (const float*)d_in[9];
    float* out = (float*)d_out;

    const int N = in_sizes[0] / 128;
    const int E = in_sizes[1] / 2;
    const int* src = ei;
    const int* dst = ei + E;

    // workspace layout (floats)
    float* ws    = (float*)d_ws;
    float* dinv  = ws;                         // N
    float* H     = ws + (long)N;               // N*64  (layer1 gemm out, then relu'd h2)
    float* AGG1  = ws + (long)N * 65;          // N*64  (layer1 agg, then layer2 gemm out)
    float* AGG2  = ws + (long)N * 129;         // N*32
    float* sums1 = ws + (long)N * 161;         // 128
    float* ab1   = sums1 + 128;                // 128
    float* sums2 = sums1 + 256;                // 64
    float* ab2   = sums2 + 64;                 // 64

    const float invN = 1.0f / (float)N;

    // 1) degrees -> dinv
    deg_init_kernel<<<cdiv_host(N, 256), 256, 0, stream>>>(dinv, N);
    deg_accum_kernel<<<cdiv_host(E, 256), 256, 0, stream>>>(dst, dinv, E);
    deg_rsqrt_kernel<<<cdiv_host(N, 256), 256, 0, stream>>>(dinv, N);

    // 2) H = x @ W1  (WMMA f32, 32 rows per block)
    gemm_wmma_kernel<128, 64><<<cdiv_host(N, 32), 256, 0, stream>>>(x, W1, H, N);

    // 3) layer-1 aggregation: self loop + bias init, then edge scatter
    agg_init_kernel<64><<<cdiv_host((long)N * 64, 256), 256, 0, stream>>>(H, dinv, b1, AGG1, N);
    edge_scatter_kernel<64><<<cdiv_host((long)E * 64, 256), 256, 0, stream>>>(H, src, dst, dinv, AGG1, E);

    // 4) BatchNorm1 + ReLU -> H (reused as h2)
    bn_zero_kernel<<<1, 256, 0, stream>>>(sums1, 128);
    bn_stats_kernel<64><<<512, 256, 0, stream>>>(AGG1, sums1, N);
    bn_final_kernel<64><<<1, 64, 0, stream>>>(sums1, g1, be1, ab1, invN);
    bn_apply_kernel<64, true, false><<<cdiv_host((long)N * 64, 256), 256, 0, stream>>>(AGG1, ab1, H, N);

    // 5) O = h2 @ W2  (WMMA f32, 64 rows per block) -> reuse AGG1 buffer
    gemm_wmma_kernel<64, 32><<<cdiv_host(N, 64), 256, 0, stream>>>(H, W2, AGG1, N);

    // 6) layer-2 aggregation
    agg_init_kernel<32><<<cdiv_host((long)N * 32, 256), 256, 0, stream>>>(AGG1, dinv, b2, AGG2, N);
    edge_scatter_kernel<32><<<cdiv_host((long)E * 32, 256), 256, 0, stream>>>(AGG1, src, dst, dinv, AGG2, E);

    // 7) BatchNorm2 (no relu) -> d_out
    bn_zero_kernel<<<1, 256, 0, stream>>>(sums2, 128);
    bn_stats_kernel<32><<<512, 256, 0, stream>>>(AGG2, sums2, N);
    bn_final_kernel<32><<<1, 64, 0, stream>>>(sums2, g2, be2, ab2, invN);
    bn_apply_kernel<32, false, true><<<cdiv_host((long)N * 32, 256), 256, 0, stream>>>(AGG2, ab2, out, N);
}